// PhysMamba_70944269795625
// MI455X (gfx1250) — compile-verified
//
#include <hip/hip_runtime.h>

// PhysMamba spectral MLP block for gfx1250 (MI455X), fp32 end-to-end.
// Pipeline: [fc1 GEMM+BN+ReLU: TDM (tensor_load_to_lds) double-buffered staging
//            + V_WMMA_F32_16X16X4_F32]
//        -> [fused FFT4096 -> per-channel complex scale+bias+ReLU -> IFFT, in LDS]
//        -> [fc2 GEMM+BN, same TDM+WMMA engine]

#define BN_EPS 1e-5f
constexpr int Bq = 8;
constexpr int Nq = 4096;
constexpr int Cq = 192;
constexpr int Hq = 768;

typedef __attribute__((ext_vector_type(2))) float        v2f;
typedef __attribute__((ext_vector_type(8))) float        v8f;
typedef __attribute__((ext_vector_type(4))) unsigned int v4u;
typedef __attribute__((ext_vector_type(8))) int          v8i;
typedef __attribute__((ext_vector_type(4))) int          v4i;

// ---------------------------------------------------------------------------
// Issue one TDM 2D tile load: tile_rows x 32 fp32 elements, row stride
// `row_stride_elems`, from global `gptr` into LDS at byte offset `lds_off`.
// D# pad fields insert 1 DWORD after every 32 DWORDs -> LDS row stride of
// 33 floats (bank-conflict padding done by the DMA engine itself).
// Issued by one wave; completion tracked on TENSORcnt.
// ---------------------------------------------------------------------------
__device__ __forceinline__ void tdm_load_tile_f32(const float* gptr,
                                                  unsigned int lds_off,
                                                  int tile_rows,
                                                  int row_stride_elems)
{
    const unsigned long long ga = (unsigned long long)(size_t)gptr;
    v4u g0;
    g0.x = 1u;                                      // count=1 valid descriptor
    g0.y = lds_off;                                 // lds_addr (bytes)
    g0.z = (unsigned int)ga;                        // global_addr[31:0]
    g0.w = (unsigned int)(ga >> 32) | (2u << 30);   // global_addr[56:32] | type=2
    v8i g1;
    // data_size=2 (4B) | pad_enable | pad_interval=4 (every 32 DW) | pad_amount=0 (1 DW)
    g1[0] = (2 << 16) | (1 << 20) | (4 << 22);
    g1[1] = 0;                                      // atomic_addr=0, tensor_dim0[15:0] (dim0=1<<20)
    g1[2] = 16;                                     // tensor_dim0[31:16]=16, tensor_dim1[15:0]=0
    g1[3] = 16 | (32 << 16);                        // tensor_dim1[31:16]=16, tile_dim0=32
    g1[4] = tile_rows;                              // tile_dim1 = rows, tile_dim2 = 0
    g1[5] = row_stride_elems;                       // tensor_dim0_stride[31:0]
    g1[6] = 0;                                      // stride0 hi / stride1 lo
    g1[7] = 0;                                      // stride1 hi
    const v4i z4 = {0, 0, 0, 0};                    // 2D tensor: groups 2/3 unused
    const v8i z8 = {0, 0, 0, 0, 0, 0, 0, 0};        // 6-arg toolchain extra group
    __builtin_amdgcn_tensor_load_to_lds(g0, g1, z4, z4, z8, 0);
}

// ---------------------------------------------------------------------------
// GEMM + BN (+optional ReLU):
//   out[row, col] = BN( sum_k A[row*K + k] * W[col*K + k] )
// A: [rows x K] row-major, W: [cols x K] row-major.
// Block = 256 threads = 8 waves; block tile 64 rows x 32 cols; each wave owns
// one 16x16 WMMA tile (4x2 wave grid). K staged in 32-wide slabs via TDM with
// ping-pong LDS buffers (TENSORcnt-pipelined, no VMEM in the hot loop).
// ---------------------------------------------------------------------------
template <int K, bool RELU>
__global__ __launch_bounds__(256) void gemm_bn_kernel(
    const float* __restrict__ A, const float* __restrict__ W,
    const float* __restrict__ gamma, const float* __restrict__ beta,
    const float* __restrict__ mean, const float* __restrict__ var,
    float* __restrict__ out, int out_ld)
{
    const int row0 = blockIdx.x * 64;
    const int col0 = blockIdx.y * 32;
    const int tid  = threadIdx.x;
    const int lane = tid & 31;
    const int wave = tid >> 5;
    const int wr   = wave & 3;   // wave row sub-tile (0..3) -> rows wr*16
    const int wc   = wave >> 2;  // wave col sub-tile (0..1) -> cols wc*16

    // TDM-padded tiles: row stride 33 floats (32 data + 1 pad DWORD).
    // sA[buf][m][k], sB[buf][n][k]; double-buffered.
    __shared__ __align__(16) float sA[2][64][33];
    __shared__ __align__(16) float sB[2][32][33];

    v8f acc = {};

    const int m16   = lane & 15;  // M (and N) index within a 16-wide tile
    const int khalf = lane >> 4;  // K split between lane halves for f32 WMMA

    constexpr int NS = K / 32;    // number of 32-wide K slabs
    const bool issuer = (wave == 0);

    // prologue: DMA slab 0 into buffer 0
    if (issuer) {
        tdm_load_tile_f32(&A[(size_t)row0 * K], (unsigned int)(size_t)&sA[0][0][0], 64, K);
        tdm_load_tile_f32(&W[(size_t)col0 * K], (unsigned int)(size_t)&sB[0][0][0], 32, K);
    }

    for (int s = 0; s < NS; s++) {
        const int buf = s & 1;
        if (issuer) {
            if (s + 1 < NS) {
                const int k1 = (s + 1) * 32;
                tdm_load_tile_f32(&A[(size_t)row0 * K + k1],
                                  (unsigned int)(size_t)&sA[buf ^ 1][0][0], 64, K);
                tdm_load_tile_f32(&W[(size_t)col0 * K + k1],
                                  (unsigned int)(size_t)&sB[buf ^ 1][0][0], 32, K);
                __builtin_amdgcn_s_wait_tensorcnt(2);  // slab s landed; s+1 in flight
            } else {
                __builtin_amdgcn_s_wait_tensorcnt(0);  // drain final slab
            }
        }
        __syncthreads();  // slab s visible to all waves

        // --- 8 WMMA ops per K-slab (K advances by 4 each) ---
#pragma unroll
        for (int kk = 0; kk < 32; kk += 4) {
            v2f av, bv;
            // A 16x4 f32: lanes 0-15 hold K={0,1}, lanes 16-31 hold K={2,3}; M = lane&15
            av.x = sA[buf][wr * 16 + m16][kk + 2 * khalf + 0];
            av.y = sA[buf][wr * 16 + m16][kk + 2 * khalf + 1];
            // B 4x16 f32: N = lane&15, K split the same way across lane halves
            bv.x = sB[buf][wc * 16 + m16][kk + 2 * khalf + 0];
            bv.y = sB[buf][wc * 16 + m16][kk + 2 * khalf + 1];
            acc = __builtin_amdgcn_wmma_f32_16x16x4_f32(
                /*neg_a=*/false, av, /*neg_b=*/false, bv,
                /*c_mod=*/(short)0, acc, /*reuse_a=*/false, /*reuse_b=*/false);
        }
        __syncthreads();  // all reads of buf done before TDM reuses it
    }

    // --- fused BatchNorm (+ReLU) epilogue ---
    // C/D layout: VGPR v holds row M = v + 8*(lane>>4), col N = lane&15.
    // All 8 accumulator entries of a lane share one output column.
    const int col = col0 + wc * 16 + m16;
    const float s = gamma[col] * rsqrtf(var[col] + BN_EPS);
    const float t = beta[col] - mean[col] * s;
#pragma unroll
    for (int v = 0; v < 8; v++) {
        const int row = row0 + wr * 16 + v + 8 * khalf;
        float o = acc[v] * s + t;
        if (RELU) o = fmaxf(o, 0.0f);
        out[(size_t)row * out_ld + col] = o;
    }
}

// ---------------------------------------------------------------------------
// Fused spectral kernel per (batch, channel):
//   FFT4096 (ortho) -> per-channel complex diag scale + bias + ReLU -> IFFT
// Forward: DIF radix-2 (natural -> bit-reversed). The pointwise op uses
// identical scalars for every bin, so bit-reversed order is harmless.
// Inverse: DIT radix-2 (bit-reversed -> natural). No permutation needed.
// All in-place in 32 KB LDS.
// ---------------------------------------------------------------------------
__global__ __launch_bounds__(256) void spectral_kernel(
    float* __restrict__ h,
    const float* __restrict__ rmat, const float* __restrict__ imat,
    const float* __restrict__ rbias, const float* __restrict__ ibias)
{
    const int ch  = blockIdx.x;  // 0..H-1
    const int bb  = blockIdx.y;  // 0..B-1
    const int tid = threadIdx.x;

    __shared__ float xr[Nq];
    __shared__ float xi[Nq];

    const size_t base = (size_t)bb * Nq * Hq + ch;
    for (int n = tid; n < Nq; n += 256) {
        xr[n] = h[base + (size_t)n * Hq];
        xi[n] = 0.0f;
    }
    __syncthreads();

    // ---- forward FFT: DIF, twiddle sign -1 ----
    for (int lh = 11; lh >= 0; lh--) {
        const int half = 1 << lh;
        const float theta = -6.28318530717958647692f / (float)(half << 1);
        for (int j = tid; j < (Nq >> 1); j += 256) {
            const int pos = j & (half - 1);
            const int i1  = ((j >> lh) << (lh + 1)) + pos;
            const int i2  = i1 + half;
            const float ur = xr[i1], ui = xi[i1];
            const float vr = xr[i2], vi = xi[i2];
            xr[i1] = ur + vr;
            xi[i1] = ui + vi;
            float sw, cw;
            __sincosf(theta * (float)pos, &sw, &cw);
            const float dr = ur - vr, di = ui - vi;
            xr[i2] = dr * cw - di * sw;
            xi[i2] = dr * sw + di * cw;
        }
        __syncthreads();
    }

    // ---- pointwise: ortho scale, complex diag mix, bias, ReLU ----
    const float rv  = rmat[(size_t)ch * Hq + ch];  // diagonal element
    const float iv  = imat[(size_t)ch * Hq + ch];
    const float rbv = rbias[ch];
    const float ibv = ibias[ch];
    const float inv = 1.0f / 64.0f;  // 1/sqrt(4096), must precede bias+ReLU
    for (int n = tid; n < Nq; n += 256) {
        const float re = xr[n] * inv;
        const float im = xi[n] * inv;
        xr[n] = fmaxf(re * rv - im * iv + rbv, 0.0f);
        xi[n] = fmaxf(im * rv + re * iv + ibv, 0.0f);
    }
    __syncthreads();

    // ---- inverse FFT: DIT, twiddle sign +1 ----
    for (int lh = 0; lh <= 11; lh++) {
        const int half = 1 << lh;
        const float theta = 6.28318530717958647692f / (float)(half << 1);
        for (int j = tid; j < (Nq >> 1); j += 256) {
            const int pos = j & (half - 1);
            const int i1  = ((j >> lh) << (lh + 1)) + pos;
            const int i2  = i1 + half;
            float sw, cw;
            __sincosf(theta * (float)pos, &sw, &cw);
            const float vr = xr[i2], vi = xi[i2];
            const float tr = vr * cw - vi * sw;
            const float ti = vr * sw + vi * cw;
            const float ur = xr[i1], ui = xi[i1];
            xr[i1] = ur + tr;
            xi[i1] = ui + ti;
            xr[i2] = ur - tr;
            xi[i2] = ui - ti;
        }
        __syncthreads();
    }

    for (int n = tid; n < Nq; n += 256) {
        h[base + (size_t)n * Hq] = xr[n] * inv;  // take real, ortho scale
    }
}

// ---------------------------------------------------------------------------
extern "C" void kernel_launch(void* const* d_in, const int* in_sizes, int n_in,
                              void* d_out, int out_size, void* d_ws, size_t ws_size,
                              hipStream_t stream)
{
    (void)in_sizes; (void)n_in; (void)out_size; (void)ws_size;
    const float* x  = (const float*)d_in[0];
    const float* W1 = (const float*)d_in[1];
    const float* g1 = (const float*)d_in[2];
    const float* b1 = (const float*)d_in[3];
    const float* m1 = (const float*)d_in[4];
    const float* v1 = (const float*)d_in[5];
    const float* r  = (const float*)d_in[6];
    const float* im = (const float*)d_in[7];
    const float* rb = (const float*)d_in[8];
    const float* ib = (const float*)d_in[9];
    const float* W2 = (const float*)d_in[10];
    const float* g2 = (const float*)d_in[11];
    const float* b2 = (const float*)d_in[12];
    const float* m2 = (const float*)d_in[13];
    const float* v2 = (const float*)d_in[14];

    float* h   = (float*)d_ws;   // [B, N, H] fp32 intermediate (96 MiB)
    float* out = (float*)d_out;  // [B, N, C]

    const int rows = Bq * Nq;  // 32768

    // fc1: x[rows x C] * W1^T -> h[rows x H], BN + ReLU
    gemm_bn_kernel<Cq, true>
        <<<dim3(rows / 64, Hq / 32), 256, 0, stream>>>(x, W1, g1, b1, m1, v1, h, Hq);

    // fused FFT -> complex diag scale + bias + ReLU -> IFFT, per (b, channel)
    spectral_kernel<<<dim3(Hq, Bq), 256, 0, stream>>>(h, r, im, rb, ib);

    // fc2: h[rows x H] * W2^T -> out[rows x C], BN
    gemm_bn_kernel<Hq, false>
        <<<dim3(rows / 64, Cq / 32), 256, 0, stream>>>(h, W2, g2, b2, m2, v2, out, Cq);
}